// SoftmaxEQL_32564442038384
// MI455X (gfx1250) — compile-verified
//
#include <hip/hip_runtime.h>
#include <hip/hip_bf16.h>
#include <math.h>

// ---------------------------------------------------------------------------
// SoftmaxEQL for MI455X (gfx1250, wave32)
//
// Roofline: logits = 32768 x 10000 fp32 = 1.31 GB >> 192MB L2, so the floor is
// one HBM pass: 1.31GB / 23.3TB/s ~= 56us. We achieve a single pass by giving
// each row to one 256-thread block and keeping the entire 40KB row resident in
// registers (10 x 16B per lane), doing max and exp-sum phases from registers.
// The 1.31GB stream has zero reuse -> non-temporal loads (TH=NT) so it does
// not thrash L2 (which keeps labels/reweighed/workspace hot).
// The exp-sum uses V_WMMA_F32_16X16X4_F32 with a ones-B matrix: per the ISA
// A-layout (16x4 f32: VGPR0 = K0/K2, VGPR1 = K1/K3 across lane halves), feeding
// (e0,e1) per lane accumulates row sums into the 8-VGPR D; D layout gives the
// full wave sum after an 8-way lane-local add + one shfl_xor(16).
// All reductions are fixed-shape (no float atomics) -> deterministic replays.
// ---------------------------------------------------------------------------

typedef __attribute__((ext_vector_type(2))) float v2f;
typedef __attribute__((ext_vector_type(4))) float v4f;
typedef __attribute__((ext_vector_type(8))) float v8f;

#define TPB 256
#define VMAX 10   // supports C <= VMAX*4*TPB = 10240 (reference C = 10000)

// ---- K1: zero histogram ----------------------------------------------------
__global__ void eql_zero_hist(int* __restrict__ hist, int C) {
    int i = blockIdx.x * blockDim.x + threadIdx.x;
    if (i < C) hist[i] = 0;
}

// ---- K2: label histogram ---------------------------------------------------
__global__ void eql_hist(const int* __restrict__ labels, int* __restrict__ hist, int B) {
    int i = blockIdx.x * blockDim.x + threadIdx.x;
    if (i < B) atomicAdd(&hist[labels[i]], 1);
}

// ---- K3: deterministic sum of (class_freqs + hist) -------------------------
__global__ __launch_bounds__(TPB) void eql_freqsum(const float* __restrict__ cf,
                                                   const int* __restrict__ hist,
                                                   float* __restrict__ freqSum, int C) {
    __shared__ float s[TPB];
    float t = 0.0f;
    for (int i = threadIdx.x; i < C; i += TPB) t += cf[i] + (float)hist[i];
    s[threadIdx.x] = t;
    __syncthreads();
    for (int off = TPB / 2; off > 0; off >>= 1) {
        if ((int)threadIdx.x < off) s[threadIdx.x] += s[threadIdx.x + off];
        __syncthreads();
    }
    if (threadIdx.x == 0) *freqSum = s[0];
}

// ---- K4: reweighed[i] = beta*(1-p) + (1-beta)*p ----------------------------
__global__ void eql_weights(const float* __restrict__ cf, const int* __restrict__ hist,
                            const float* __restrict__ beta, const float* __restrict__ freqSum,
                            float* __restrict__ rw, int C) {
    int i = blockIdx.x * blockDim.x + threadIdx.x;
    if (i < C) {
        float p = (cf[i] + (float)hist[i]) / (*freqSum);
        float b = beta[i];
        // b*(1-p) + (1-b)*p = b + p - 2*b*p
        rw[i] = fmaf(-2.0f * b, p, b + p);
    }
}

// ---- K5: main per-row kernel (single HBM pass, NT loads, WMMA exp-sum) -----
__global__ __launch_bounds__(TPB) void eql_row_kernel(const float* __restrict__ logits,
                                                      const int* __restrict__ labels,
                                                      const float* __restrict__ rw,
                                                      float* __restrict__ rowNum,
                                                      float* __restrict__ rowDen, int C) {
    const int row  = blockIdx.x;
    const int tid  = threadIdx.x;
    const int lane = tid & 31;
    const int wave = tid >> 5;
    const size_t base = (size_t)row * (size_t)C;
    const v4f* rowp = (const v4f*)(logits + base);
    const int nV4 = C >> 2;

    // Phase 1: stream the row into registers (non-temporal: single-use data),
    // tracking the running max.
    v4f vals[VMAX];
    float m = -INFINITY;
#pragma unroll
    for (int i = 0; i < VMAX; ++i) {
        int idx = tid + i * TPB;
        v4f v;
        if (idx < nV4) {
            v = __builtin_nontemporal_load(&rowp[idx]);   // global_load_b128 th:NT
        } else {
            v = (v4f)(-INFINITY);                          // exp(-inf - M) = 0 later
        }
        vals[i] = v;
        m = fmaxf(m, fmaxf(fmaxf(v.x, v.y), fmaxf(v.z, v.w)));
    }

    // Wave32 max reduce, then block max via LDS.
    for (int off = 16; off > 0; off >>= 1) m = fmaxf(m, __shfl_xor(m, off, 32));
    __shared__ float sred[TPB / 32];
    __shared__ float sbc;
    if (lane == 0) sred[wave] = m;
    __syncthreads();
    if (wave == 0) {
        float mm = (lane < (TPB / 32)) ? sred[lane] : -INFINITY;
        for (int off = 16; off > 0; off >>= 1) mm = fmaxf(mm, __shfl_xor(mm, off, 32));
        if (lane == 0) sbc = mm;
    }
    __syncthreads();
    const float M = sbc;

    // Phase 2: exp from registers, accumulate Sum(exp) with WMMA f32 16x16x4,
    // B = ones so D[m][n] += A[m][k] for this wave's values. EXEC is all-1s
    // here (no divergence around the WMMA ops).
    v8f acc = {};
    const v2f ones = {1.0f, 1.0f};
#pragma unroll
    for (int i = 0; i < VMAX; ++i) {
        v4f v = vals[i];
        float e0 = __expf(v.x - M);
        float e1 = __expf(v.y - M);
        float e2 = __expf(v.z - M);
        float e3 = __expf(v.w - M);
        v2f a0 = {e0, e1};
        acc = __builtin_amdgcn_wmma_f32_16x16x4_f32(false, a0, false, ones,
                                                    (short)0, acc, false, false);
        v2f a1 = {e2, e3};
        acc = __builtin_amdgcn_wmma_f32_16x16x4_f32(false, a1, false, ones,
                                                    (short)0, acc, false, false);
    }
    // D VGPR j holds row M=j (lanes 0-15) / M=j+8 (lanes 16-31): lane-local
    // 8-add gives half-wave sum; shfl_xor(16) add completes the wave sum.
    float part = acc[0] + acc[1] + acc[2] + acc[3] + acc[4] + acc[5] + acc[6] + acc[7];
    float wsum = part + __shfl_xor(part, 16, 32);

    __syncthreads();               // sred reads above are complete; safe to reuse
    if (lane == 0) sred[wave] = wsum;
    __syncthreads();
    if (tid == 0) {
        float S = 0.0f;
#pragma unroll
        for (int wv = 0; wv < TPB / 32; ++wv) S += sred[wv];   // fixed order
        int lab   = labels[row];
        float xl  = logits[base + (size_t)lab];
        float w   = rw[lab];
        float nll = -(xl - M - logf(S));
        rowNum[row] = w * nll;
        rowDen[row] = w;
    }
}

// ---- K6: deterministic final reduction + divide ----------------------------
__global__ __launch_bounds__(TPB) void eql_final(const float* __restrict__ rowNum,
                                                 const float* __restrict__ rowDen,
                                                 float* __restrict__ out, int B) {
    __shared__ float sn[TPB];
    __shared__ float sd[TPB];
    float tn = 0.0f, td = 0.0f;
    for (int i = threadIdx.x; i < B; i += TPB) { tn += rowNum[i]; td += rowDen[i]; }
    sn[threadIdx.x] = tn;
    sd[threadIdx.x] = td;
    __syncthreads();
    for (int off = TPB / 2; off > 0; off >>= 1) {
        if ((int)threadIdx.x < off) {
            sn[threadIdx.x] += sn[threadIdx.x + off];
            sd[threadIdx.x] += sd[threadIdx.x + off];
        }
        __syncthreads();
    }
    if (threadIdx.x == 0) out[0] = sn[0] / sd[0];
}

// ---------------------------------------------------------------------------
extern "C" void kernel_launch(void* const* d_in, const int* in_sizes, int n_in,
                              void* d_out, int out_size, void* d_ws, size_t ws_size,
                              hipStream_t stream) {
    const float* logits = (const float*)d_in[0];   // [B, C] f32
    const int*   labels = (const int*)d_in[1];     // [B]    int
    const float* cfreqs = (const float*)d_in[2];   // [C]    f32
    const float* beta   = (const float*)d_in[3];   // [C]    f32
    float* out = (float*)d_out;

    const int B = in_sizes[1];
    const int C = in_sizes[2];

    // Workspace layout (all 4-byte elements):
    //   [0, C)        hist (int)
    //   [C, 2C)       reweighed (float)
    //   [2C, 2C+B)    rowNum (float)
    //   [2C+B, 2C+2B) rowDen (float)
    //   [2C+2B]       freqSum (float)
    int*   hist    = (int*)d_ws;
    float* rw      = (float*)d_ws + C;
    float* rowNum  = (float*)d_ws + 2 * (size_t)C;
    float* rowDen  = rowNum + B;
    float* freqSum = rowDen + B;

    const int gC = (C + TPB - 1) / TPB;
    const int gB = (B + TPB - 1) / TPB;

    eql_zero_hist<<<gC, TPB, 0, stream>>>(hist, C);
    eql_hist<<<gB, TPB, 0, stream>>>(labels, hist, B);
    eql_freqsum<<<1, TPB, 0, stream>>>(cfreqs, hist, freqSum, C);
    eql_weights<<<gC, TPB, 0, stream>>>(cfreqs, hist, beta, freqSum, rw, C);
    eql_row_kernel<<<B, TPB, 0, stream>>>(logits, labels, rw, rowNum, rowDen, C);
    eql_final<<<1, TPB, 0, stream>>>(rowNum, rowDen, out, B);
}